// BiModel_14723147891241
// MI455X (gfx1250) — compile-verified
//
#include <hip/hip_runtime.h>
#include <hip/hip_bf16.h>
#include <stdint.h>

#define NNODES 50000
#define NEDGES 800000
#define F_IN   500
#define HID    64
#define COUT   16

typedef __attribute__((ext_vector_type(2))) float v2f;
typedef __attribute__((ext_vector_type(8))) float v8f;

// ---------------------------------------------------------------- utilities
__global__ void fill_zero_kernel(float* __restrict__ p, int n) {
    int i = blockIdx.x * blockDim.x + threadIdx.x;
    if (i < n) p[i] = 0.0f;
}

// ------------------------------------------------------------- degree pass
// deg_st[d] += (1-rev), deg_ts[d] += rev   (self-loop +1 added later)
__global__ void edge_degree_kernel(const int* __restrict__ dst,
                                   const unsigned char* __restrict__ rev,
                                   float* __restrict__ deg_st,
                                   float* __restrict__ deg_ts) {
    int e = blockIdx.x * blockDim.x + threadIdx.x;
    if (e >= NEDGES) return;
    int d = dst[e];
    if (rev[e]) atomicAdd(&deg_ts[d], 1.0f);
    else        atomicAdd(&deg_st[d], 1.0f);
}

// dinv = rsqrt(deg + 1);  deg_all = deg_st + deg_ts + 1
__global__ void dinv_kernel(const float* __restrict__ deg_st,
                            const float* __restrict__ deg_ts,
                            float* __restrict__ dinv_st,
                            float* __restrict__ dinv_ts,
                            float* __restrict__ dinv_all) {
    int i = blockIdx.x * blockDim.x + threadIdx.x;
    if (i >= NNODES) return;
    float s = deg_st[i], t = deg_ts[i];
    dinv_st[i]  = rsqrtf(s + 1.0f);
    dinv_ts[i]  = rsqrtf(t + 1.0f);
    dinv_all[i] = rsqrtf(s + t + 1.0f);
}

// ------------------------------------------------------------- f32 WMMA GEMM
// C[M x N] = A[M x K] @ W[K x N], written at C[m*ldc + col0 + n].
// One wave computes a 16 x (NT*16) strip via V_WMMA_F32_16X16X4_F32.
// A fragment (16x4 f32): lanes 0-15 hold rows M=lane, K={k,k+1};
//                        lanes 16-31 hold rows M=lane-16, K={k+2,k+3}.
// B fragment (4x16 f32): lanes 0-15 hold cols N=lane, K={k,k+1};
//                        lanes 16-31 hold cols N=lane-16, K={k+2,k+3}.
// D (16x16 f32): VGPR r, lanes 0-15 -> M=r, N=lane; lanes 16-31 -> M=8+r.
template <int NT>
__global__ void gemm_wmma_f32_kernel(const float* __restrict__ A,
                                     const float* __restrict__ W,
                                     float* __restrict__ C,
                                     int M, int K, int N, int ldc, int col0) {
    const int wave = threadIdx.x >> 5;
    const int lane = threadIdx.x & 31;
    const int wavesPerBlock = blockDim.x >> 5;
    const int row0 = (blockIdx.x * wavesPerBlock + wave) * 16;
    if (row0 >= M) return;                       // wave-uniform: EXEC stays full

    const int half = lane >> 4;                  // 0: K lo pair, 1: K hi pair
    const int l    = lane & 15;

    v8f acc[NT];
#pragma unroll
    for (int t = 0; t < NT; ++t) {
        v8f z = {0.f, 0.f, 0.f, 0.f, 0.f, 0.f, 0.f, 0.f};
        acc[t] = z;
    }

    const float* arow = A + (size_t)(row0 + l) * K;
    for (int k = 0; k < K; k += 4) {
        v2f a = *(const v2f*)(arow + k + 2 * half);      // 8B aligned by construction
        const float* wk0 = W + (size_t)(k + 2 * half) * N;
#pragma unroll
        for (int t = 0; t < NT; ++t) {
            int n = t * 16 + l;
            v2f b;
            b.x = wk0[n];
            b.y = wk0[N + n];
            acc[t] = __builtin_amdgcn_wmma_f32_16x16x4_f32(
                false, a, false, b, (short)0, acc[t], false, false);
        }
    }

#pragma unroll
    for (int t = 0; t < NT; ++t) {
#pragma unroll
        for (int r = 0; r < 8; ++r) {
            int m = row0 + r + 8 * half;
            C[(size_t)m * ldc + col0 + t * 16 + l] = acc[t][r];
        }
    }
}

// ------------------------------------------------- edge aggregation (dual branch)
// xw layout: [N,128] (cols 0..63 = st branch, 64..127 = ts branch).
// Each edge belongs to exactly one branch (masks are complementary).
// 16 threads/edge, float4 per thread.
__global__ void edge_agg_dual_kernel(const int* __restrict__ src,
                                     const int* __restrict__ dst,
                                     const unsigned char* __restrict__ rev,
                                     const float* __restrict__ dinv_st,
                                     const float* __restrict__ dinv_ts,
                                     const float* __restrict__ xw,
                                     float* __restrict__ agg) {
    long long tid = (long long)blockIdx.x * blockDim.x + threadIdx.x;
    int e = (int)(tid >> 4);
    int q = (int)(tid & 15);
    if (e >= NEDGES) return;
    int s = src[e], d = dst[e];
    int r = rev[e] != 0;
    const float* dinv = r ? dinv_ts : dinv_st;
    float coef = dinv[s] * dinv[d];
    int base = r ? HID : 0;
    const float4 v = *(const float4*)(xw + (size_t)s * 128 + base + q * 4);
    float* out = agg + (size_t)d * 128 + base + q * 4;
    atomicAdd(out + 0, v.x * coef);
    atomicAdd(out + 1, v.y * coef);
    atomicAdd(out + 2, v.z * coef);
    atomicAdd(out + 3, v.w * coef);
}

// h[i,c] = relu(agg[i,c] + h[i,c]*dinv_b(i)^2 + bias_b[c]) for both branches (in place)
__global__ void combine_relu_kernel(float* __restrict__ h,
                                    const float* __restrict__ agg,
                                    const float* __restrict__ dinv_st,
                                    const float* __restrict__ dinv_ts,
                                    const float* __restrict__ b_st,
                                    const float* __restrict__ b_ts) {
    long long tid = (long long)blockIdx.x * blockDim.x + threadIdx.x;
    if (tid >= (long long)NNODES * 128) return;
    int i = (int)(tid >> 7);
    int c = (int)(tid & 127);
    int ts = c >= HID;
    float di = ts ? dinv_ts[i] : dinv_st[i];
    float bi = ts ? b_ts[c - HID] : b_st[c];
    float v = agg[tid] + h[tid] * (di * di) + bi;
    h[tid] = fmaxf(v, 0.0f);
}

// ------------------------------------------------- last-layer aggregation (all edges)
// 4 threads/edge, float4 per thread over 16 channels.
__global__ void edge_agg_all_kernel(const int* __restrict__ src,
                                    const int* __restrict__ dst,
                                    const float* __restrict__ dinv_all,
                                    const float* __restrict__ xw16,
                                    float* __restrict__ agg16) {
    long long tid = (long long)blockIdx.x * blockDim.x + threadIdx.x;
    int e = (int)(tid >> 2);
    int q = (int)(tid & 3);
    if (e >= NEDGES) return;
    int s = src[e], d = dst[e];
    float coef = dinv_all[s] * dinv_all[d];
    const float4 v = *(const float4*)(xw16 + (size_t)s * COUT + q * 4);
    float* out = agg16 + (size_t)d * COUT + q * 4;
    atomicAdd(out + 0, v.x * coef);
    atomicAdd(out + 1, v.y * coef);
    atomicAdd(out + 2, v.z * coef);
    atomicAdd(out + 3, v.w * coef);
}

// out[i,:] = log_softmax(agg16[i,:] + xw16[i,:]*dinv_all[i]^2 + b_last)
__global__ void softmax_kernel(const float* __restrict__ xw16,
                               const float* __restrict__ agg16,
                               const float* __restrict__ dinv_all,
                               const float* __restrict__ b_last,
                               float* __restrict__ out) {
    int i = blockIdx.x * blockDim.x + threadIdx.x;
    if (i >= NNODES) return;
    float di = dinv_all[i];
    float d2 = di * di;
    float v[COUT];
    float mx = -3.4e38f;
#pragma unroll
    for (int c = 0; c < COUT; ++c) {
        v[c] = agg16[(size_t)i * COUT + c] + xw16[(size_t)i * COUT + c] * d2 + b_last[c];
        mx = fmaxf(mx, v[c]);
    }
    float sum = 0.0f;
#pragma unroll
    for (int c = 0; c < COUT; ++c) sum += __expf(v[c] - mx);
    float lse = mx + __logf(sum);
#pragma unroll
    for (int c = 0; c < COUT; ++c) out[(size_t)i * COUT + c] = v[c] - lse;
}

static inline int cdiv(long long a, int b) { return (int)((a + b - 1) / b); }

extern "C" void kernel_launch(void* const* d_in, const int* in_sizes, int n_in,
                              void* d_out, int out_size, void* d_ws, size_t ws_size,
                              hipStream_t stream) {
    const float* x       = (const float*)d_in[0];                 // [N, F_IN]
    const int*   edge    = (const int*)d_in[1];                   // [2, E] flat
    const unsigned char* rev = (const unsigned char*)d_in[2];     // [E] bool
    const float* W_st0 = (const float*)d_in[3];
    const float* b_st0 = (const float*)d_in[4];
    const float* W_ts0 = (const float*)d_in[5];
    const float* b_ts0 = (const float*)d_in[6];
    const float* W_st1 = (const float*)d_in[7];
    const float* b_st1 = (const float*)d_in[8];
    const float* W_ts1 = (const float*)d_in[9];
    const float* b_ts1 = (const float*)d_in[10];
    const float* W_last = (const float*)d_in[11];
    const float* b_last = (const float*)d_in[12];
    float* out = (float*)d_out;

    const int* src = edge;
    const int* dst = edge + NEDGES;

    // ---- workspace layout (floats) ----
    float* ws = (float*)d_ws;
    float* deg_st   = ws;                 ws += NNODES;
    float* deg_ts   = ws;                 ws += NNODES;
    float* dinv_st  = ws;                 ws += NNODES;
    float* dinv_ts  = ws;                 ws += NNODES;
    float* dinv_all = ws;                 ws += NNODES;
    float* h1  = ws;                      ws += (size_t)NNODES * 128;
    float* h2  = ws;                      ws += (size_t)NNODES * 128;
    float* agg = ws;                      ws += (size_t)NNODES * 128;
    float* xw16  = agg;                   // reuse agg region after layer 1
    float* agg16 = agg + (size_t)NNODES * COUT;

    const int B = 256;

    // ---- degrees and normalization coefficients ----
    fill_zero_kernel<<<cdiv(2 * NNODES, B), B, 0, stream>>>(deg_st, 2 * NNODES);
    edge_degree_kernel<<<cdiv(NEDGES, B), B, 0, stream>>>(dst, rev, deg_st, deg_ts);
    dinv_kernel<<<cdiv(NNODES, B), B, 0, stream>>>(deg_st, deg_ts, dinv_st, dinv_ts, dinv_all);

    const int gemmBlocks64 = cdiv(NNODES / 16, 4);  // 4 waves/block, 16 rows/wave

    // ---- layer 0: xw into concat layout of h1 ----
    gemm_wmma_f32_kernel<4><<<gemmBlocks64, 128, 0, stream>>>(x, W_st0, h1, NNODES, F_IN, HID, 128, 0);
    gemm_wmma_f32_kernel<4><<<gemmBlocks64, 128, 0, stream>>>(x, W_ts0, h1, NNODES, F_IN, HID, 128, HID);
    fill_zero_kernel<<<cdiv((long long)NNODES * 128, B), B, 0, stream>>>(agg, NNODES * 128);
    edge_agg_dual_kernel<<<cdiv((long long)NEDGES * 16, B), B, 0, stream>>>(src, dst, rev, dinv_st, dinv_ts, h1, agg);
    combine_relu_kernel<<<cdiv((long long)NNODES * 128, B), B, 0, stream>>>(h1, agg, dinv_st, dinv_ts, b_st0, b_ts0);

    // ---- layer 1: h1 [N,128] -> h2 [N,128] ----
    gemm_wmma_f32_kernel<4><<<gemmBlocks64, 128, 0, stream>>>(h1, W_st1, h2, NNODES, 128, HID, 128, 0);
    gemm_wmma_f32_kernel<4><<<gemmBlocks64, 128, 0, stream>>>(h1, W_ts1, h2, NNODES, 128, HID, 128, HID);
    fill_zero_kernel<<<cdiv((long long)NNODES * 128, B), B, 0, stream>>>(agg, NNODES * 128);
    edge_agg_dual_kernel<<<cdiv((long long)NEDGES * 16, B), B, 0, stream>>>(src, dst, rev, dinv_st, dinv_ts, h2, agg);
    combine_relu_kernel<<<cdiv((long long)NNODES * 128, B), B, 0, stream>>>(h2, agg, dinv_st, dinv_ts, b_st1, b_ts1);

    // ---- last layer: h2 @ W_last -> xw16 [N,16]; aggregate over all edges ----
    gemm_wmma_f32_kernel<1><<<gemmBlocks64, 128, 0, stream>>>(h2, W_last, xw16, NNODES, 128, COUT, COUT, 0);
    fill_zero_kernel<<<cdiv((long long)NNODES * COUT, B), B, 0, stream>>>(agg16, NNODES * COUT);
    edge_agg_all_kernel<<<cdiv((long long)NEDGES * 4, B), B, 0, stream>>>(src, dst, dinv_all, xw16, agg16);
    softmax_kernel<<<cdiv(NNODES, B), B, 0, stream>>>(xw16, agg16, dinv_all, b_last, out);
}